// AttnDecoderRNN_32976758899258
// MI455X (gfx1250) — compile-verified
//
#include <hip/hip_runtime.h>
#include <hip/hip_bf16.h>
#include <math.h>

// Problem sizes (match reference)
#define Hs 1024
#define Vs 50257
#define Ls 128

typedef __attribute__((ext_vector_type(2))) float v2f;
typedef __attribute__((ext_vector_type(4))) float v4f;
typedef __attribute__((ext_vector_type(8))) float v8f;

// ---------------------------------------------------------------------------
// WMMA-based 16-row GEMV tile:
//   acc[m] += dot(W[row0+m, 0:K], xvec[0:K])   for m = 0..15
// using V_WMMA_F32_16X16X4_F32 (D = A x B + C).
//
// K-permuted mapping (dot products are K-order invariant): per 8-wide K chunk,
// lanes 0-15 own physical K {k..k+3} (one contiguous float4 from their row),
// lanes 16-31 own {k+4..k+7}.  WMMA #1 consumes (q.xy) with B chunk (x.xy) at
// the same physical K; WMMA #2 consumes (q.zw)/(x.zw).  Every A slot pairs
// with the B value of the same physical k, so D accumulates the exact dot
// product.  All 16 B columns are identical (broadcast), so every column of D
// holds the 16 row results.
//
// Per 8 K: 1 global_load_b128 + 1 ds_load_b128 + 2 wmma.  Two accumulators
// break the C-chain RAW dependency for ILP.
// Requires: whole wave active, K % 8 == 0, row0 % 16 == 0, xvec 16B-aligned.
// ---------------------------------------------------------------------------
__device__ __forceinline__ v8f gemv16_wmma(const float* __restrict__ W, int ldk,
                                           int row0, int maxrow,
                                           const float* xvec, int K, v8f acc) {
    const int lane = threadIdx.x & 31;
    const int m    = lane & 15;
    const int kq   = (lane >> 4) << 2;          // 0 for lanes 0-15, 4 for 16-31
    int r = row0 + m;
    if (r > maxrow) r = maxrow;                  // clamp (keeps EXEC full)
    const float* wp = W + (size_t)r * (size_t)ldk + kq;
    const float* xp = xvec + kq;
    v8f acc1 = {};
#pragma unroll 4
    for (int k = 0; k < K; k += 8) {
        v4f a = *(const v4f*)(wp + k);           // global_load_b128 per lane
        v4f b = *(const v4f*)(xp + k);           // ds_load_b128 broadcast
        v2f a0 = {a.x, a.y}, a1 = {a.z, a.w};
        v2f b0 = {b.x, b.y}, b1 = {b.z, b.w};
        acc  = __builtin_amdgcn_wmma_f32_16x16x4_f32(
                   false, a0, false, b0, (short)0, acc,  false, false);
        acc1 = __builtin_amdgcn_wmma_f32_16x16x4_f32(
                   false, a1, false, b1, (short)0, acc1, false, false);
    }
#pragma unroll
    for (int i = 0; i < 8; ++i) acc[i] += acc1[i];
    return acc;
}

// D-matrix layout: VGPR v, lanes 0-15 -> M=v ; lanes 16-31 -> M=v+8 (all N
// columns identical).  Lane 0 holds rows row0..row0+7, lane 16 rows +8..+15.

// ---------------------------------------------------------------------------
// 1) Attention scores: logits[l] = dot(attn_W[l,:], [emb | h0]) + attn_b[l]
//    rows = 128, K = 2H.  1 block x 256 threads = 8 waves = 8 row-tiles.
// ---------------------------------------------------------------------------
__global__ void attn_scores_kernel(const int* __restrict__ tokens,
                                   const float* __restrict__ emb,
                                   const float* __restrict__ h0,
                                   const float* __restrict__ attn_W,
                                   const float* __restrict__ attn_b,
                                   float* __restrict__ logits) {
    __shared__ __align__(16) float xv[2 * Hs];
    const int tok = tokens[0];
    for (int i = threadIdx.x; i < Hs; i += blockDim.x) {
        xv[i]      = emb[(size_t)tok * Hs + i];
        xv[Hs + i] = h0[i];
    }
    __syncthreads();
    const int wave = threadIdx.x >> 5;           // 0..7
    const int row0 = wave * 16;                  // 0..112
    v8f acc = {};
    acc = gemv16_wmma(attn_W, 2 * Hs, row0, Ls - 1, xv, 2 * Hs, acc);
    const int lane = threadIdx.x & 31;
    if ((lane & 15) == 0) {
        const int base = row0 + ((lane >> 4) << 3);
#pragma unroll
        for (int i = 0; i < 8; ++i)
            logits[base + i] = acc[i] + attn_b[base + i];
    }
}

// ---------------------------------------------------------------------------
// 2) Softmax over 128 attention logits (1 block x 128 threads)
// ---------------------------------------------------------------------------
__global__ void softmax128_kernel(const float* __restrict__ logits,
                                  float* __restrict__ w) {
    __shared__ float red[Ls];
    const int t = threadIdx.x;
    const float v = logits[t];
    red[t] = v;
    __syncthreads();
    for (int s = Ls / 2; s > 0; s >>= 1) {
        if (t < s) red[t] = fmaxf(red[t], red[t + s]);
        __syncthreads();
    }
    const float mx = red[0];
    __syncthreads();
    const float e = __expf(v - mx);
    red[t] = e;
    __syncthreads();
    for (int s = Ls / 2; s > 0; s >>= 1) {
        if (t < s) red[t] += red[t + s];
        __syncthreads();
    }
    w[t] = e / red[0];
}

// ---------------------------------------------------------------------------
// 3) attn_applied[j] = sum_l w[l] * enc[l, j]   (coalesced column reduction)
// ---------------------------------------------------------------------------
__global__ void attn_apply_kernel(const float* __restrict__ w,
                                  const float* __restrict__ enc,
                                  float* __restrict__ applied) {
    __shared__ float wl[Ls];
    if (threadIdx.x < Ls) wl[threadIdx.x] = w[threadIdx.x];
    __syncthreads();
    const int j = blockIdx.x * blockDim.x + threadIdx.x;   // 0..H-1
    float s = 0.f;
#pragma unroll 8
    for (int l = 0; l < Ls; ++l)
        s += wl[l] * enc[(size_t)l * Hs + j];
    applied[j] = s;
}

// ---------------------------------------------------------------------------
// 4) Combine: x[r] = relu(dot(comb_W[r,:], [emb | applied]) + comb_b[r])
//    rows = H, K = 2H.  8 blocks x 256 = 64 waves.
// ---------------------------------------------------------------------------
__global__ void combine_kernel(const int* __restrict__ tokens,
                               const float* __restrict__ emb,
                               const float* __restrict__ applied,
                               const float* __restrict__ comb_W,
                               const float* __restrict__ comb_b,
                               float* __restrict__ x) {
    __shared__ __align__(16) float xv[2 * Hs];
    const int tok = tokens[0];
    for (int i = threadIdx.x; i < Hs; i += blockDim.x) {
        xv[i]      = emb[(size_t)tok * Hs + i];
        xv[Hs + i] = applied[i];
    }
    __syncthreads();
    const int gwave = blockIdx.x * (blockDim.x >> 5) + (threadIdx.x >> 5);
    const int row0  = gwave * 16;
    v8f acc = {};
    acc = gemv16_wmma(comb_W, 2 * Hs, row0, Hs - 1, xv, 2 * Hs, acc);
    const int lane = threadIdx.x & 31;
    if ((lane & 15) == 0) {
        const int base = row0 + ((lane >> 4) << 3);
#pragma unroll
        for (int i = 0; i < 8; ++i)
            x[base + i] = fmaxf(acc[i] + comb_b[base + i], 0.f);
    }
}

// ---------------------------------------------------------------------------
// 5) LSTM gates: g[r] = dot(W_ih[r,:],x) + dot(W_hh[r,:],h0) + b_ih[r]+b_hh[r]
//    rows = 4H, K = H (x2 matrices into one accumulator). 32 blocks x 256.
// ---------------------------------------------------------------------------
__global__ void gates_kernel(const float* __restrict__ xin,
                             const float* __restrict__ h0,
                             const float* __restrict__ W_ih,
                             const float* __restrict__ W_hh,
                             const float* __restrict__ b_ih,
                             const float* __restrict__ b_hh,
                             float* __restrict__ gates) {
    __shared__ __align__(16) float xv[Hs];
    __shared__ __align__(16) float hv[Hs];
    for (int i = threadIdx.x; i < Hs; i += blockDim.x) {
        xv[i] = xin[i];
        hv[i] = h0[i];
    }
    __syncthreads();
    const int gwave = blockIdx.x * (blockDim.x >> 5) + (threadIdx.x >> 5);
    const int row0  = gwave * 16;
    v8f acc = {};
    acc = gemv16_wmma(W_ih, Hs, row0, 4 * Hs - 1, xv, Hs, acc);
    acc = gemv16_wmma(W_hh, Hs, row0, 4 * Hs - 1, hv, Hs, acc);
    const int lane = threadIdx.x & 31;
    if ((lane & 15) == 0) {
        const int base = row0 + ((lane >> 4) << 3);
#pragma unroll
        for (int i = 0; i < 8; ++i)
            gates[base + i] = acc[i] + b_ih[base + i] + b_hh[base + i];
    }
}

// ---------------------------------------------------------------------------
// 6) LSTM pointwise (PyTorch gate order i,f,g,o)
// ---------------------------------------------------------------------------
__global__ void lstm_kernel(const float* __restrict__ gates,
                            const float* __restrict__ c0,
                            float* __restrict__ hnew,
                            float* __restrict__ cnew) {
    const int j = blockIdx.x * blockDim.x + threadIdx.x;
    const float ig = gates[j];
    const float fg = gates[Hs + j];
    const float gg = gates[2 * Hs + j];
    const float og = gates[3 * Hs + j];
    const float si = 1.f / (1.f + __expf(-ig));
    const float sf = 1.f / (1.f + __expf(-fg));
    const float so = 1.f / (1.f + __expf(-og));
    const float c  = sf * c0[j] + si * tanhf(gg);
    cnew[j] = c;
    hnew[j] = so * tanhf(c);
}

// ---------------------------------------------------------------------------
// 7) Output projection: logits[r] = dot(out_W[r,:], h_new) + out_b[r]
//    rows = V = 50257 (206 MB => the bandwidth-dominant kernel).
//    ceil(V/16)=3142 row-tiles, 8 waves/block.
// ---------------------------------------------------------------------------
__global__ void outproj_kernel(const float* __restrict__ hnew,
                               const float* __restrict__ out_W,
                               const float* __restrict__ out_b,
                               float* __restrict__ logits) {
    __shared__ __align__(16) float hv[Hs];
    for (int i = threadIdx.x; i < Hs; i += blockDim.x) hv[i] = hnew[i];
    __syncthreads();
    const int gwave = blockIdx.x * (blockDim.x >> 5) + (threadIdx.x >> 5);
    const int row0  = gwave * 16;
    if (row0 >= Vs) return;                       // wave-uniform exit
    v8f acc = {};
    acc = gemv16_wmma(out_W, Hs, row0, Vs - 1, hv, Hs, acc);
    const int lane = threadIdx.x & 31;
    if ((lane & 15) == 0) {
        const int base = row0 + ((lane >> 4) << 3);
#pragma unroll
        for (int i = 0; i < 8; ++i) {
            const int r = base + i;
            if (r < Vs) logits[r] = acc[i] + out_b[r];
        }
    }
}

// ---------------------------------------------------------------------------
// 8) In-place log_softmax over V logits (1 block x 1024 threads, 3 passes in L2)
// ---------------------------------------------------------------------------
__global__ void logsoftmax_kernel(float* __restrict__ x, int n) {
    __shared__ float red[1024];
    const int t = threadIdx.x;
    float mx = -INFINITY;
    for (int i = t; i < n; i += 1024) mx = fmaxf(mx, x[i]);
    red[t] = mx;
    __syncthreads();
    for (int s = 512; s > 0; s >>= 1) {
        if (t < s) red[t] = fmaxf(red[t], red[t + s]);
        __syncthreads();
    }
    mx = red[0];
    __syncthreads();
    float sum = 0.f;
    for (int i = t; i < n; i += 1024) sum += __expf(x[i] - mx);
    red[t] = sum;
    __syncthreads();
    for (int s = 512; s > 0; s >>= 1) {
        if (t < s) red[t] += red[t + s];
        __syncthreads();
    }
    const float lse = mx + logf(red[0]);
    for (int i = t; i < n; i += 1024) x[i] -= lse;
}

// ---------------------------------------------------------------------------
extern "C" void kernel_launch(void* const* d_in, const int* in_sizes, int n_in,
                              void* d_out, int out_size, void* d_ws, size_t ws_size,
                              hipStream_t stream) {
    const int*   tokens = (const int*)  d_in[0];
    const float* h0     = (const float*)d_in[1];
    const float* c0     = (const float*)d_in[2];
    const float* enc    = (const float*)d_in[3];
    const float* emb    = (const float*)d_in[4];
    const float* attn_W = (const float*)d_in[5];
    const float* attn_b = (const float*)d_in[6];
    const float* comb_W = (const float*)d_in[7];
    const float* comb_b = (const float*)d_in[8];
    const float* W_ih   = (const float*)d_in[9];
    const float* W_hh   = (const float*)d_in[10];
    const float* b_ih   = (const float*)d_in[11];
    const float* b_hh   = (const float*)d_in[12];
    const float* out_W  = (const float*)d_in[13];
    const float* out_b  = (const float*)d_in[14];

    // Output layout (return order): log_softmax[V] | h_new[H] | c_new[H] | attn_w[L]
    float* out      = (float*)d_out;
    float* o_logits = out;
    float* o_h      = out + Vs;
    float* o_c      = out + Vs + Hs;
    float* o_attnw  = out + Vs + 2 * Hs;

    // Workspace (floats): attn logits | attn_applied | x | gates  (~25 KB)
    float* ws         = (float*)d_ws;
    float* ws_logits  = ws;                 // 128
    float* ws_applied = ws + Ls;            // 1024
    float* ws_x       = ws + Ls + Hs;       // 1024
    float* ws_gates   = ws + Ls + 2 * Hs;   // 4096

    attn_scores_kernel<<<1, 256, 0, stream>>>(tokens, emb, h0, attn_W, attn_b, ws_logits);
    softmax128_kernel<<<1, Ls, 0, stream>>>(ws_logits, o_attnw);
    attn_apply_kernel<<<Hs / 256, 256, 0, stream>>>(o_attnw, enc, ws_applied);
    combine_kernel<<<(Hs / 16) / 8, 256, 0, stream>>>(tokens, emb, ws_applied,
                                                      comb_W, comb_b, ws_x);
    gates_kernel<<<(4 * Hs / 16) / 8, 256, 0, stream>>>(ws_x, h0, W_ih, W_hh,
                                                        b_ih, b_hh, ws_gates);
    lstm_kernel<<<Hs / 256, 256, 0, stream>>>(ws_gates, c0, o_h, o_c);

    const int vtiles  = (Vs + 15) / 16;          // 3142
    const int vblocks = (vtiles + 7) / 8;        // 393
    outproj_kernel<<<vblocks, 256, 0, stream>>>(o_h, out_W, out_b, o_logits);
    logsoftmax_kernel<<<1, 1024, 0, stream>>>(o_logits, Vs);
}